// FourierMMLayer_86002425135413
// MI455X (gfx1250) — compile-verified
//
#include <hip/hip_runtime.h>

// FNet Fourier layer: out = Re(F_S @ x @ F_H) = C_S@(x@C_H) - S_S@(x@S_H)
// 4 real f32 GEMMs on the CDNA5 WMMA pipe (V_WMMA_F32_16X16X4_F32).
// B operands staged per-block into double-buffered LDS via the gfx1250 async
// global->LDS engine (GLOBAL_LOAD_ASYNC_TO_LDS_B128 / ASYNCcnt); B fragments
// assembled with hand-paired ds_load_2addr_b32 (one s_wait_dscnt per 8 loads).

typedef float v2f __attribute__((ext_vector_type(2)));
typedef float v8f __attribute__((ext_vector_type(8)));

constexpr int BATCH = 2;
constexpr int SEQ   = 2048;
constexpr int HID   = 768;
constexpr int MROWS = BATCH * SEQ;   // 4096 flattened rows of x
constexpr int NSTR  = HID / 64;      // 12 column strips of 64
constexpr int KC    = 32;            // K-chunk staged in LDS per block
constexpr unsigned BUFB = KC * 64 * 4;  // LDS bytes per buffer plane (8 KB)

// ---- gfx1250 async global->LDS copy (16B per lane), ASYNCcnt-tracked ------
__device__ __forceinline__ void async_copy16(unsigned lds_off, const void* g) {
    asm volatile("global_load_async_to_lds_b128 %0, %1, off"
                 :: "v"(lds_off), "v"((unsigned long long)(uintptr_t)g)
                 : "memory");
}
__device__ __forceinline__ void async_wait0() {
    asm volatile("s_wait_asynccnt 0" ::: "memory");
}

// ---- 8 correctly-paired B fragments (4 col-tiles x 2 matrices) ------------
// Each ds_load_2addr_b32 drops {row k, row k+1} of one 16-col tile into a
// 64-bit VGPR pair == one v2f WMMA B fragment. One dscnt wait for all 8.
__device__ __forceinline__ void ds_load_frag8(
    unsigned a1, unsigned a2,
    v2f& b10, v2f& b11, v2f& b12, v2f& b13,
    v2f& b20, v2f& b21, v2f& b22, v2f& b23) {
    asm volatile(
        "ds_load_2addr_b32 %0, %8 offset0:0 offset1:64\n\t"
        "ds_load_2addr_b32 %1, %8 offset0:16 offset1:80\n\t"
        "ds_load_2addr_b32 %2, %8 offset0:32 offset1:96\n\t"
        "ds_load_2addr_b32 %3, %8 offset0:48 offset1:112\n\t"
        "ds_load_2addr_b32 %4, %9 offset0:0 offset1:64\n\t"
        "ds_load_2addr_b32 %5, %9 offset0:16 offset1:80\n\t"
        "ds_load_2addr_b32 %6, %9 offset0:32 offset1:96\n\t"
        "ds_load_2addr_b32 %7, %9 offset0:48 offset1:112\n\t"
        "s_wait_dscnt 0"
        : "=&v"(b10), "=&v"(b11), "=&v"(b12), "=&v"(b13),
          "=&v"(b20), "=&v"(b21), "=&v"(b22), "=&v"(b23)
        : "v"(a1), "v"(a2));
}

// ---------------------------------------------------------------------------
// Split complex128 hidden-DFT into planar f32 Re / Im planes.
// ---------------------------------------------------------------------------
__global__ void planarize_hidden(const double* __restrict__ dftH,
                                 float* __restrict__ Ch,
                                 float* __restrict__ Sh) {
    int i = blockIdx.x * blockDim.x + threadIdx.x;
    if (i < HID * HID) {
        Ch[i] = (float)dftH[2 * i];
        Sh[i] = (float)dftH[2 * i + 1];
    }
}

// ---------------------------------------------------------------------------
// GEMM1 (fused pair): T1 = x @ Ch, T2 = x @ Sh      (M=4096, N=768, K=768)
// Block: 8 waves -> 128x64 tile; Ch/Sh chunks double-buffered in LDS.
// A-frag (f32 16x4): lane&15 = M row; lanes 0-15 hold K=k,k+1; 16-31 K=k+2,k+3.
// B-frag (f32 4x16): lane&15 = N col; v[0] row = k+(lane>>4)*2, v[1] = +1.
// ---------------------------------------------------------------------------
__global__ __launch_bounds__(256) void gemm_hidden(
    const float* __restrict__ X,      // [4096, 768]
    const float* __restrict__ Ch,     // [768, 768]
    const float* __restrict__ Sh,     // [768, 768]
    float* __restrict__ T1,           // [4096, 768]
    float* __restrict__ T2) {
    __shared__ float bs1[2][KC][64];
    __shared__ float bs2[2][KC][64];

    const int wv   = threadIdx.x >> 5;
    const int lane = threadIdx.x & 31;
    const int mt   = blockIdx.x / NSTR;        // 0..31 (128-row block tiles)
    const int ns   = blockIdx.x % NSTR;        // 0..11
    const int m0   = mt * 128 + wv * 16;
    const int n0   = ns * 64;
    const int row  = lane & 15;
    const int half = lane >> 4;
    const int koff = half * 2;

    const float* arow = X + (size_t)(m0 + row) * HID + koff;
    const unsigned l1 = (unsigned)(uintptr_t)&bs1[0][0][0];
    const unsigned l2 = (unsigned)(uintptr_t)&bs2[0][0][0];
    const unsigned fragoff = (unsigned)((koff * 64 + row) * 4);
    const int er = threadIdx.x >> 4;           // staging row within chunk half
    const int ec = (threadIdx.x & 15) * 4;     // staging col (16B aligned)

    auto stage = [&](int buf, int k0) {
#pragma unroll
        for (int i = 0; i < 2; ++i) {
            const int r = er + i * 16;         // 0..31
            const size_t g = (size_t)(k0 + r) * HID + n0 + ec;
            const unsigned lo = (unsigned)buf * BUFB + (unsigned)((r * 64 + ec) * 4);
            async_copy16(l1 + lo, Ch + g);
            async_copy16(l2 + lo, Sh + g);
        }
    };

    v8f c1[4] = {}, c2[4] = {};

    stage(0, 0);
    async_wait0();
    __syncthreads();

    int buf = 0;
    for (int k0 = 0; k0 < HID; k0 += KC) {
        if (k0 + KC < HID) stage(buf ^ 1, k0 + KC);   // overlap with compute
        const unsigned base1 = l1 + (unsigned)buf * BUFB + fragoff;
        const unsigned base2 = l2 + (unsigned)buf * BUFB + fragoff;

#pragma unroll
        for (int kk = 0; kk < KC; kk += 4) {
            v2f a = *(const v2f*)(arow + k0 + kk);
            v2f b10, b11, b12, b13, b20, b21, b22, b23;
            ds_load_frag8(base1 + (unsigned)(kk * 256),
                          base2 + (unsigned)(kk * 256),
                          b10, b11, b12, b13, b20, b21, b22, b23);
            c1[0] = __builtin_amdgcn_wmma_f32_16x16x4_f32(false, a, false, b10, (short)0, c1[0], false, false);
            c1[1] = __builtin_amdgcn_wmma_f32_16x16x4_f32(false, a, false, b11, (short)0, c1[1], false, false);
            c1[2] = __builtin_amdgcn_wmma_f32_16x16x4_f32(false, a, false, b12, (short)0, c1[2], false, false);
            c1[3] = __builtin_amdgcn_wmma_f32_16x16x4_f32(false, a, false, b13, (short)0, c1[3], false, false);
            c2[0] = __builtin_amdgcn_wmma_f32_16x16x4_f32(false, a, false, b20, (short)0, c2[0], false, false);
            c2[1] = __builtin_amdgcn_wmma_f32_16x16x4_f32(false, a, false, b21, (short)0, c2[1], false, false);
            c2[2] = __builtin_amdgcn_wmma_f32_16x16x4_f32(false, a, false, b22, (short)0, c2[2], false, false);
            c2[3] = __builtin_amdgcn_wmma_f32_16x16x4_f32(false, a, false, b23, (short)0, c2[3], false, false);
        }
        async_wait0();       // next chunk's copies landed during compute
        __syncthreads();
        buf ^= 1;
    }

#pragma unroll
    for (int t = 0; t < 4; ++t) {
#pragma unroll
        for (int v = 0; v < 8; ++v) {
            const size_t r = (size_t)(m0 + v + half * 8) * HID + n0 + t * 16 + row;
            T1[r] = c1[t][v];
            T2[r] = c2[t][v];
        }
    }
}

// ---------------------------------------------------------------------------
// GEMM2 (fused pair): out[b] = Cs @ T1[b] + (-Ss) @ T2[b]
//   M=2048, N=768, K=2048 per batch. T1/T2 chunks double-buffered in LDS;
//   A fragments built on the fly from complex128 (one double2 = Re & Im).
// ---------------------------------------------------------------------------
__global__ __launch_bounds__(256) void gemm_seq(
    const double* __restrict__ dftS,  // [2048, 2048] complex128 interleaved
    const float* __restrict__ T1,     // [4096, 768]
    const float* __restrict__ T2,
    float* __restrict__ out) {        // [4096, 768]
    __shared__ float bs1[2][KC][64];
    __shared__ float bs2[2][KC][64];

    const int wv   = threadIdx.x >> 5;
    const int lane = threadIdx.x & 31;
    const int MT   = SEQ / 128;                        // 16 block tiles per batch
    const int b    = blockIdx.x / (MT * NSTR);         // 0..1
    const int rem  = blockIdx.x % (MT * NSTR);
    const int mt   = rem / NSTR;                       // 0..15
    const int ns   = rem % NSTR;                       // 0..11
    const int m0   = mt * 128 + wv * 16;
    const int n0   = ns * 64;
    const int row  = lane & 15;
    const int half = lane >> 4;
    const int koff = half * 2;

    const double2* arow = (const double2*)dftS + (size_t)(m0 + row) * SEQ + koff;
    const float*   t1b  = T1 + (size_t)b * SEQ * HID;
    const float*   t2b  = T2 + (size_t)b * SEQ * HID;
    const unsigned l1 = (unsigned)(uintptr_t)&bs1[0][0][0];
    const unsigned l2 = (unsigned)(uintptr_t)&bs2[0][0][0];
    const unsigned fragoff = (unsigned)((koff * 64 + row) * 4);
    const int er = threadIdx.x >> 4;
    const int ec = (threadIdx.x & 15) * 4;

    auto stage = [&](int buf, int k0) {
#pragma unroll
        for (int i = 0; i < 2; ++i) {
            const int r = er + i * 16;
            const size_t g = (size_t)(k0 + r) * HID + n0 + ec;
            const unsigned lo = (unsigned)buf * BUFB + (unsigned)((r * 64 + ec) * 4);
            async_copy16(l1 + lo, t1b + g);
            async_copy16(l2 + lo, t2b + g);
        }
    };

    v8f c[4] = {};

    stage(0, 0);
    async_wait0();
    __syncthreads();

    int buf = 0;
    for (int k0 = 0; k0 < SEQ; k0 += KC) {
        if (k0 + KC < SEQ) stage(buf ^ 1, k0 + KC);   // overlap with compute
        const unsigned base1 = l1 + (unsigned)buf * BUFB + fragoff;
        const unsigned base2 = l2 + (unsigned)buf * BUFB + fragoff;

#pragma unroll
        for (int kk = 0; kk < KC; kk += 4) {
            double2 e0 = arow[k0 + kk];        // (Re, Im) at K = k0+kk+koff
            double2 e1 = arow[k0 + kk + 1];
            v2f ac, as;
            ac.x = (float)e0.x;  ac.y = (float)e1.x;   // cos plane (Re)
            as.x = -(float)e0.y; as.y = -(float)e1.y;  // -Im plane
            v2f b10, b11, b12, b13, b20, b21, b22, b23;
            ds_load_frag8(base1 + (unsigned)(kk * 256),
                          base2 + (unsigned)(kk * 256),
                          b10, b11, b12, b13, b20, b21, b22, b23);
            c[0] = __builtin_amdgcn_wmma_f32_16x16x4_f32(false, ac, false, b10, (short)0, c[0], false, false);
            c[0] = __builtin_amdgcn_wmma_f32_16x16x4_f32(false, as, false, b20, (short)0, c[0], false, false);
            c[1] = __builtin_amdgcn_wmma_f32_16x16x4_f32(false, ac, false, b11, (short)0, c[1], false, false);
            c[1] = __builtin_amdgcn_wmma_f32_16x16x4_f32(false, as, false, b21, (short)0, c[1], false, false);
            c[2] = __builtin_amdgcn_wmma_f32_16x16x4_f32(false, ac, false, b12, (short)0, c[2], false, false);
            c[2] = __builtin_amdgcn_wmma_f32_16x16x4_f32(false, as, false, b22, (short)0, c[2], false, false);
            c[3] = __builtin_amdgcn_wmma_f32_16x16x4_f32(false, ac, false, b13, (short)0, c[3], false, false);
            c[3] = __builtin_amdgcn_wmma_f32_16x16x4_f32(false, as, false, b23, (short)0, c[3], false, false);
        }
        async_wait0();
        __syncthreads();
        buf ^= 1;
    }

#pragma unroll
    for (int t = 0; t < 4; ++t) {
#pragma unroll
        for (int v = 0; v < 8; ++v) {
            const size_t r = (size_t)(b * SEQ + m0 + v + half * 8) * HID
                           + n0 + t * 16 + row;
            out[r] = c[t][v];
        }
    }
}

// ---------------------------------------------------------------------------
extern "C" void kernel_launch(void* const* d_in, const int* in_sizes, int n_in,
                              void* d_out, int out_size, void* d_ws, size_t ws_size,
                              hipStream_t stream) {
    const float*  X    = (const float*)d_in[0];   // [2,2048,768] f32
    const double* dftS = (const double*)d_in[1];  // [2048,2048] complex128
    const double* dftH = (const double*)d_in[2];  // [768,768]  complex128
    float* out = (float*)d_out;

    // Workspace layout (~30 MB): T1 | T2 | Ch | Sh
    float* T1 = (float*)d_ws;
    float* T2 = T1 + (size_t)MROWS * HID;
    float* Ch = T2 + (size_t)MROWS * HID;
    float* Sh = Ch + (size_t)HID * HID;

    planarize_hidden<<<(HID * HID + 255) / 256, 256, 0, stream>>>(dftH, Ch, Sh);

    // 384 blocks x 256 threads each (exact tiling, no partial tiles).
    gemm_hidden<<<384, 256, 0, stream>>>(X, Ch, Sh, T1, T2);
    gemm_seq<<<384, 256, 0, stream>>>(dftS, T1, T2, out);
}